// SophisticatedRouter_11166914970138
// MI455X (gfx1250) — compile-verified
//
#include <hip/hip_runtime.h>
#include <cstdint>

// ---------------------------------------------------------------------------
// Expert-choice router for MI455X (gfx1250).
//   N=262144 tokens, E=64 experts, K=capacity=4096, outputs:
//     [0          .. E*K)   top-k token indices (int32 bits)
//     [E*K        .. 2*E*K) normalized expert weights (f32)
//     [2*E*K      .. 3*E*K) expert mask (all 1.0f)
// Pipeline (memory-bound, ~260MB total traffic ≈ 11us @ 23.3TB/s):
//   K1: TDM tile load -> LDS (padded stride-65 layout via TDM pad feature),
//       softmax per token (row max + v_exp_f32), row sums via
//       v_wmma_f32_16x16x4_f32 with a ones B-matrix, transposed store of
//       probs to ws scoresT[E][N].
//   K2: per-expert 4096-bin histogram of float bits >> 19 (LDS-aggregated).
//   K3: suffix scan of histogram -> threshold bin, count_above, needed ties.
//   K4: compaction into exactly K slots per expert (atomic append).
//   K5: in-LDS bitonic sort of 4096 (val,idx) pairs, desc value / asc index,
//       sum-normalize, write all three outputs.
// ---------------------------------------------------------------------------

#define N_TOKENS 262144
#define N_EXPERTS 64
#define CAP 4096
#define BINS 4096
#define BIN_SHIFT 19
#define TILE_T 128          // tokens per block in K1
#define TILE_STRIDE 65      // padded LDS row stride (conflict-free)

typedef float v2f __attribute__((ext_vector_type(2)));
typedef float v8f __attribute__((ext_vector_type(8)));
typedef unsigned int v4u __attribute__((ext_vector_type(4)));
typedef int v4i __attribute__((ext_vector_type(4)));
typedef int v8i __attribute__((ext_vector_type(8)));

// ---- workspace layout (bytes) ----
#define OFF_SCORES 0u                       // float[E][N]        64 MB
#define OFF_HIST   67108864u                // uint[E][BINS]       1 MB
#define OFF_META   68157440u                // int[320] (5x64)
#define OFF_PVAL   68161536u                // float[E][CAP]       1 MB
#define OFF_PIDX   69210112u                // int[E][CAP]         1 MB
// meta sub-arrays (ints): [0..63] thresh_bin, [64..127] count_above,
// [128..191] needed, [192..255] above_ctr, [256..319] tie_ctr

// ===========================================================================
// K1: softmax + transpose. One block = 128 tokens x 64 experts.
// ===========================================================================
__global__ __launch_bounds__(256) void k1_softmax_transpose(
    const float* __restrict__ logits, float* __restrict__ scoresT) {
  __shared__ float tile[TILE_T * TILE_STRIDE];  // 33280 B, padded rows
  __shared__ float sums[TILE_T];
  const int tid = threadIdx.x;

  // ---- TDM: DMA the 128x64 f32 tile into LDS with 1-DWORD pad per 64 ----
  if (tid < 32) {  // single wave issues the tensor load (EXEC ignored by TDM)
    uint64_t ga = (uint64_t)(uintptr_t)logits + (uint64_t)blockIdx.x * (TILE_T * 64 * 4);
    uint32_t la = (uint32_t)(uintptr_t)(&tile[0]);
    v4u g0;
    g0.x = 1u;                                   // count=1 valid descriptor
    g0.y = la;                                   // lds_addr (bytes)
    g0.z = (uint32_t)ga;                         // global_addr[31:0]
    g0.w = (uint32_t)(ga >> 32) | (2u << 30);    // global_addr[56:32] | type=2
    // group1: data_size=4B(2), pad_enable, pad_interval=64 DWORDs (code 5),
    // pad_amount=1 DWORD (code 0); 1-D tile of 8192 elements.
    v8i g1 = { (int)0x01520000,
               (int)(8192 << 16),    // tensor_dim0 = 8192 (low16 @ [31:16])
               (int)(1 << 16),       // tensor_dim1 = 1
               (int)(8192 << 16),    // tile_dim0   = 8192
               1,                    // tile_dim1   = 1
               8192,                 // tensor_dim0_stride
               0, 0 };
    v4i z4 = {0, 0, 0, 0};
    v8i z8 = {0, 0, 0, 0, 0, 0, 0, 0};
    __builtin_amdgcn_tensor_load_to_lds(g0, g1, z4, z4, z8, 0);
    __builtin_amdgcn_s_wait_tensorcnt(0);
  }
  __syncthreads();

  // ---- stable softmax numerator per token (one thread per token) ----
  if (tid < TILE_T) {
    float m = -1e30f;
#pragma unroll 8
    for (int e = 0; e < 64; ++e) m = fmaxf(m, tile[tid * TILE_STRIDE + e]);
#pragma unroll 8
    for (int e = 0; e < 64; ++e)
      tile[tid * TILE_STRIDE + e] = __expf(tile[tid * TILE_STRIDE + e] - m);
  }
  __syncthreads();

  // ---- row sums via V_WMMA_F32_16X16X4_F32: D = A(16x4) * ones(4x16) ----
  // Wave w covers tokens w*16 .. w*16+15; 16 WMMA steps cover K=64 experts.
  {
    const int wave = tid >> 5, lane = tid & 31;
    const int M = lane & 15, hi = lane >> 4;   // A: lanes0-15 K0/K1, 16-31 K2/K3
    const int tok = wave * 16 + M;
    v8f acc = {};
    v2f ones; ones.x = 1.0f; ones.y = 1.0f;
#pragma unroll
    for (int c = 0; c < 16; ++c) {
      v2f a;
      a.x = tile[tok * TILE_STRIDE + c * 4 + hi * 2];
      a.y = tile[tok * TILE_STRIDE + c * 4 + hi * 2 + 1];
      acc = __builtin_amdgcn_wmma_f32_16x16x4_f32(
          false, a, false, ones, (short)0, acc, false, false);
    }
    // D: VGPR r, lane-half h holds token (r + 8*h); every column equal.
    if (M == 0) {
#pragma unroll
      for (int r = 0; r < 8; ++r) sums[wave * 16 + r + hi * 8] = acc[r];
    }
  }
  __syncthreads();
  if (tid < TILE_T) sums[tid] = 1.0f / sums[tid];
  __syncthreads();

  // ---- transposed, coalesced store of probabilities ----
  const size_t tokBase = (size_t)blockIdx.x * TILE_T;
#pragma unroll
  for (int it = 0; it < 32; ++it) {
    int idx = it * 256 + tid;          // 0..8191
    int e = idx >> 7, t = idx & 127;   // consecutive tid -> consecutive t
    scoresT[(size_t)e * N_TOKENS + tokBase + t] =
        tile[t * TILE_STRIDE + e] * sums[t];
  }
}

// ===========================================================================
// K2: per-expert histogram, LDS-aggregated. grid = (64 experts, 8 chunks).
// ===========================================================================
__global__ __launch_bounds__(256) void k2_histogram(
    const float* __restrict__ scoresT, unsigned* __restrict__ hist) {
  __shared__ unsigned lh[BINS];
  const int e = blockIdx.x, tid = threadIdx.x;
  for (int b = tid; b < BINS; b += 256) lh[b] = 0u;
  __syncthreads();
  const size_t base = (size_t)e * N_TOKENS + (size_t)blockIdx.y * 32768;
#pragma unroll 4
  for (int i = 0; i < 128; ++i) {
    float v = scoresT[base + i * 256 + tid];
    atomicAdd(&lh[__float_as_uint(v) >> BIN_SHIFT], 1u);
  }
  __syncthreads();
  for (int b = tid; b < BINS; b += 256)
    if (lh[b]) atomicAdd(&hist[e * BINS + b], lh[b]);
}

// ===========================================================================
// K3: threshold selection (suffix scan from the top bin).
// ===========================================================================
__global__ void k3_select(const unsigned* __restrict__ hist,
                          int* __restrict__ meta) {
  const int e = blockIdx.x;
  if (threadIdx.x == 0) {
    unsigned acc = 0;
    for (int b = BINS - 1; b >= 0; --b) {
      unsigned h = hist[e * BINS + b];
      if (acc + h >= (unsigned)CAP) {
        meta[e] = b;                        // threshold bin
        meta[64 + e] = (int)acc;            // strictly-above count
        meta[128 + e] = CAP - (int)acc;     // ties needed from this bin
        return;
      }
      acc += h;
    }
  }
}

// ===========================================================================
// K4: compaction into exactly CAP slots per expert.
// ===========================================================================
__global__ __launch_bounds__(256) void k4_compact(
    const float* __restrict__ scoresT, int* __restrict__ meta,
    float* __restrict__ pv, int* __restrict__ pi) {
  const int e = blockIdx.x, tid = threadIdx.x;
  const int tb = meta[e], cAbove = meta[64 + e], needed = meta[128 + e];
  const int base = blockIdx.y * 32768;
#pragma unroll 4
  for (int i = 0; i < 128; ++i) {
    int t = base + i * 256 + tid;
    float v = scoresT[(size_t)e * N_TOKENS + t];
    int bin = (int)(__float_as_uint(v) >> BIN_SHIFT);
    int slot = -1;
    if (bin > tb) {
      slot = atomicAdd(&meta[192 + e], 1);
    } else if (bin == tb) {
      int q = atomicAdd(&meta[256 + e], 1);
      if (q < needed) slot = cAbove + q;
    }
    if (slot >= 0) { pv[e * CAP + slot] = v; pi[e * CAP + slot] = t; }
  }
}

// ===========================================================================
// K5: bitonic sort (desc value, asc index), normalize, emit outputs.
// ===========================================================================
__global__ __launch_bounds__(256) void k5_sort_out(
    const float* __restrict__ pv, const int* __restrict__ pi,
    float* __restrict__ out) {
  __shared__ float sv[CAP];
  __shared__ int si[CAP];
  __shared__ float red[256];
  const int e = blockIdx.x, tid = threadIdx.x;
  for (int i = tid; i < CAP; i += 256) { sv[i] = pv[e * CAP + i]; si[i] = pi[e * CAP + i]; }
  __syncthreads();

  for (int k = 2; k <= CAP; k <<= 1) {
    for (int j = k >> 1; j > 0; j >>= 1) {
      for (int i = tid; i < CAP; i += 256) {
        int ixj = i ^ j;
        if (ixj > i) {
          float va = sv[i], vb = sv[ixj];
          int ia = si[i], ib = si[ixj];
          bool up = ((i & k) == 0);
          // aPrec: element i should precede ixj in final order (desc val, asc idx)
          bool aPrec = (va > vb) || (va == vb && ia < ib);
          if (aPrec != up) { sv[i] = vb; sv[ixj] = va; si[i] = ib; si[ixj] = ia; }
        }
      }
      __syncthreads();
    }
  }

  // sum of selected scores
  float s = 0.0f;
  for (int i = tid; i < CAP; i += 256) s += sv[i];
  red[tid] = s;
  __syncthreads();
  for (int off = 128; off > 0; off >>= 1) {
    if (tid < off) red[tid] += red[tid + off];
    __syncthreads();
  }
  const float inv = 1.0f / (red[0] + 1e-8f);

  int* outIdx = (int*)out;                       // int32 bits, first segment
  float* outW = out + N_EXPERTS * CAP;
  float* outM = out + 2 * N_EXPERTS * CAP;
  for (int i = tid; i < CAP; i += 256) {
    outIdx[e * CAP + i] = si[i];
    outW[e * CAP + i] = sv[i] * inv;
    outM[e * CAP + i] = 1.0f;
  }
}

// ===========================================================================
extern "C" void kernel_launch(void* const* d_in, const int* in_sizes, int n_in,
                              void* d_out, int out_size, void* d_ws, size_t ws_size,
                              hipStream_t stream) {
  (void)in_sizes; (void)n_in; (void)out_size; (void)ws_size;
  const float* logits = (const float*)d_in[0];   // [N, 64] f32
  float* out = (float*)d_out;
  uint8_t* ws = (uint8_t*)d_ws;

  float* scoresT = (float*)(ws + OFF_SCORES);
  unsigned* hist = (unsigned*)(ws + OFF_HIST);
  int* meta = (int*)(ws + OFF_META);
  float* pval = (float*)(ws + OFF_PVAL);
  int* pidx = (int*)(ws + OFF_PIDX);

  // zero histogram + meta/counters (graph-capture legal)
  (void)hipMemsetAsync(hist, 0,
                       (size_t)N_EXPERTS * BINS * sizeof(unsigned) + 4096, stream);

  k1_softmax_transpose<<<N_TOKENS / TILE_T, 256, 0, stream>>>(logits, scoresT);
  k2_histogram<<<dim3(N_EXPERTS, 8), 256, 0, stream>>>(scoresT, hist);
  k3_select<<<N_EXPERTS, 32, 0, stream>>>(hist, meta);
  k4_compact<<<dim3(N_EXPERTS, 8), 256, 0, stream>>>(scoresT, meta, pval, pidx);
  k5_sort_out<<<N_EXPERTS, 256, 0, stream>>>(pval, pidx, out);
}